// ConvGraph_19645180412611
// MI455X (gfx1250) — compile-verified
//
#include <hip/hip_runtime.h>

// Problem constants (match reference)
#define BATCH   2
#define NVERT   20000
#define NEDGE   3
#define NNBR    16
#define FDIM    128          // F_IN == UNITS
#define KDIM    (NEDGE*FDIM) // 384
#define ROWS    32           // vertices per block (2 WMMA M-tiles)
#define XSTR    388          // LDS row stride (384 + 4 pad -> bank-conflict-free column reads)
#define NBLK    ((BATCH*NVERT)/ROWS) // 1250
#define KSTEPS  (KDIM/4)     // 96 wmma k-steps
#define NTASK   (ROWS*NEDGE) // 96 gather tasks per block

typedef float v2f __attribute__((ext_vector_type(2)));
typedef float v8f __attribute__((ext_vector_type(8)));

// ---------------------------------------------------------------------------
// Pack kernels (E,F,U) row-major -> per-lane WMMA B fragments.
// Fragment order: [kstep 0..95][ntile 0..7][lane 0..31] -> float2 {K=k, K=k+1}
// with k = kstep*4 + 2*(lane>=16), col = ntile*16 + (lane&15).
// Lane-contiguous float2 => fully coalesced 256B loads in the GEMM.
// ---------------------------------------------------------------------------
__global__ __launch_bounds__(256) void pack_b_kernel(const float* __restrict__ K,
                                                     float* __restrict__ Bp) {
    int idx = blockIdx.x * 256 + threadIdx.x;      // one float2 per thread
    if (idx >= KSTEPS * 8 * 32) return;
    int ks   = idx >> 8;          // / 256
    int rem  = idx & 255;
    int nt   = rem >> 5;
    int lane = rem & 31;
    int hi   = lane >> 4;
    int col  = nt * 16 + (lane & 15);
    int k    = ks * 4 + 2 * hi;
    Bp[idx * 2 + 0] = K[(k + 0) * FDIM + col];
    Bp[idx * 2 + 1] = K[(k + 1) * FDIM + col];
}

// ---------------------------------------------------------------------------
// Fused gather/mean + WMMA GEMM + epilogue.
// Block handles 32 consecutive global rows (gv = b*V + v); V%32==0 so the
// batch index is uniform per block.
// ---------------------------------------------------------------------------
__global__ __launch_bounds__(128) void conv_graph_kernel(
    const int*   __restrict__ adj,    // (B,V,E,N)
    const float* __restrict__ feats,  // (B,V,F)
    const v2f*   __restrict__ Bp,     // packed B fragments
    const float* __restrict__ bias,   // (E,U)
    float*       __restrict__ out) {  // (B,V,U)

    __shared__ float Xs[ROWS * XSTR];           // 49.7 KB aggregated features
    __shared__ int   adjS[ROWS * NEDGE * NNBR]; // 6 KB
    __shared__ int   emaskS[ROWS];

    const int tid        = threadIdx.x;
    const int lane       = tid & 31;
    const int w          = tid >> 5;            // wave 0..3
    const int blockStart = blockIdx.x * ROWS;
    const int bb         = (blockStart >= NVERT) ? 1 : 0;
    const float* fb      = feats + (size_t)bb * NVERT * FDIM;

    // ---- stage adjacency (contiguous across the 32 rows) ----
    const int abase = blockStart * (NEDGE * NNBR);
    for (int i = tid; i < ROWS * NEDGE * NNBR; i += 128)
        adjS[i] = adj[abase + i];
    __syncthreads();

    // ---- per-row edge-type presence mask ----
    if (tid < ROWS) {
        int m = 0;
        #pragma unroll
        for (int e = 0; e < NEDGE; ++e) {
            int any = 0;
            #pragma unroll
            for (int n = 0; n < NNBR; ++n)
                any |= (adjS[tid * 48 + e * 16 + n] >= 0);
            m |= any << e;
        }
        emaskS[tid] = m;
    }

    // ---- gather + mean into X tile ----
    // Wave w owns tasks [w*24, w*24+24); each lane covers 4 channels so every
    // neighbour-row fetch is one coalesced b128 load (512B per wave).
    const int ch = lane * 4;
    for (int t = w * (NTASK / 4); t < (w + 1) * (NTASK / 4); ++t) {
        const int r = t / NEDGE;
        const int e = t - r * NEDGE;
        const int* an = &adjS[r * 48 + e * 16];
        float4 acc = make_float4(0.f, 0.f, 0.f, 0.f);
        int cnt = 0;
        #pragma unroll
        for (int n = 0; n < NNBR; ++n) {
            int id = an[n];                       // uniform across the wave
            if (id >= 0) {
                cnt++;
                const float4 f = *(const float4*)&fb[(size_t)id * FDIM + ch];
                acc.x += f.x; acc.y += f.y; acc.z += f.z; acc.w += f.w;
            }
        }
        const float deg = (float)(cnt > 0 ? cnt : 1);
        float4 o = make_float4(acc.x / deg, acc.y / deg, acc.z / deg, acc.w / deg);
        *(float4*)&Xs[r * XSTR + e * FDIM + ch] = o;
    }
    __syncthreads();

    // ---- WMMA GEMM: 16x384 @ 384x128 per M-tile, exact f32 ----
    const int hi  = lane >> 4;    // lane half selects K pair
    const int l16 = lane & 15;

    v8f c00 = {}, c01 = {}, c10 = {}, c11 = {};

    #pragma unroll 4
    for (int ks = 0; ks < KSTEPS; ++ks) {
        v2f b0 = Bp[(ks * 8 + 2 * w + 0) * 32 + lane];
        v2f b1 = Bp[(ks * 8 + 2 * w + 1) * 32 + lane];
        int kk = ks * 4 + 2 * hi;
        v2f a0, a1;
        a0.x = Xs[l16 * XSTR + kk];
        a0.y = Xs[l16 * XSTR + kk + 1];
        a1.x = Xs[(16 + l16) * XSTR + kk];
        a1.y = Xs[(16 + l16) * XSTR + kk + 1];
        c00 = __builtin_amdgcn_wmma_f32_16x16x4_f32(false, a0, false, b0, (short)0, c00, false, false);
        c01 = __builtin_amdgcn_wmma_f32_16x16x4_f32(false, a0, false, b1, (short)0, c01, false, false);
        c10 = __builtin_amdgcn_wmma_f32_16x16x4_f32(false, a1, false, b0, (short)0, c10, false, false);
        c11 = __builtin_amdgcn_wmma_f32_16x16x4_f32(false, a1, false, b1, (short)0, c11, false, false);
    }

    // ---- branchless epilogue: gated bias + relu + isolated-vertex fallback ----
    // C/D layout: VGPR i -> (M=i, N=lane) lanes 0-15, (M=i+8, N=lane-16) lanes 16-31
    auto epi = [&](const v8f& c, int mt, int nt) {
        const int col = nt * 16 + l16;
        const float b0 = bias[0 * FDIM + col];
        const float b1 = bias[1 * FDIM + col];
        const float b2 = bias[2 * FDIM + col];
        #pragma unroll
        for (int i = 0; i < 8; ++i) {
            const int r  = mt * 16 + i + 8 * hi;
            const int gv = blockStart + r;
            const int m  = emaskS[r];
            float bs = ((m & 1) ? b0 : 0.f) + ((m & 2) ? b1 : 0.f) + ((m & 4) ? b2 : 0.f);
            float pooled   = fmaxf(c[i] + bs, 0.f);
            float fallback = feats[(size_t)gv * FDIM + col];
            out[(size_t)gv * FDIM + col] = m ? pooled : fallback;
        }
    };
    epi(c00, 0, 2 * w);
    epi(c01, 0, 2 * w + 1);
    epi(c10, 1, 2 * w);
    epi(c11, 1, 2 * w + 1);
}

extern "C" void kernel_launch(void* const* d_in, const int* in_sizes, int n_in,
                              void* d_out, int out_size, void* d_ws, size_t ws_size,
                              hipStream_t stream) {
    const int*   adj   = (const int*)  d_in[0];   // (B,V,E,N) int32
    const float* feats = (const float*)d_in[1];   // (B,V,F)   f32
    const float* kers  = (const float*)d_in[2];   // (E,F,U)   f32  == stacked (384,128)
    const float* bias  = (const float*)d_in[3];   // (E,U)     f32
    float*       out   = (float*)d_out;           // (B,V,U)   f32
    float*       Bp    = (float*)d_ws;            // 196608 B packed B fragments

    // 1) pack kernel matrix into WMMA B-fragment order (24576 float2)
    pack_b_kernel<<<(KSTEPS * 8 * 32 + 255) / 256, 256, 0, stream>>>(kers, Bp);

    // 2) fused gather/mean + GEMM + epilogue
    conv_graph_kernel<<<NBLK, 128, 0, stream>>>(adj, feats, (const v2f*)Bp, bias, out);
}